// Scalar_37984690766340
// MI455X (gfx1250) — compile-verified
//
#include <hip/hip_runtime.h>

// Embedding gather: out[row, :] = W[idx[row], :] + b  (BATCH=65536, EMBED=16)
// Pure bandwidth-bound gather; W (64 MB) fits in MI455X's 192 MB global L2.
// 4 lanes cooperate on one 64-byte embedding row, one b128 load each.

typedef __attribute__((ext_vector_type(4))) float v4f;

#define BATCH 65536

__global__ __launch_bounds__(256) void embed_gather_kernel(
    const int* __restrict__ idx,   // [BATCH] (stored as [BATCH,1])
    const float* __restrict__ W,   // [VOCAB, 16]
    const float* __restrict__ b,   // [16]
    float* __restrict__ out)       // [BATCH, 16]
{
    // Grid exactly covers BATCH*4 threads: no bounds guard needed.
    const unsigned tid = blockIdx.x * 256u + threadIdx.x; // 0 .. BATCH*4-1
    const unsigned row = tid >> 2;   // embedding row this lane helps with
    const unsigned sub = tid & 3;    // which float4 of the 64 B row

    // Index stream read once -> non-temporal so it doesn't pollute L2
    // (where the 64 MB W table stays resident across graph replays).
    const int r = __builtin_nontemporal_load(idx + row);

    // All byte offsets fit in 32 bits (W spans 64 MB, out spans 4 MB):
    // keep address math in one 32-bit VGPR so loads/stores use the
    // GVS mode (SGPR64 base + VGPR32 offset) instead of 64-bit VALU adds.
    const unsigned woff = ((unsigned)r << 6) | (sub << 4); // r*64 + sub*16 bytes

    const v4f w = *reinterpret_cast<const v4f*>(
        reinterpret_cast<const char*>(W) + woff);           // regular-temporal: L2-resident
    const v4f bias = reinterpret_cast<const v4f*>(b)[sub];  // 64 B, L0-resident
    const v4f o = w + bias;                                 // v_pk_add_f32 x2

    // Output written once, never re-read by the kernel -> NT store.
    __builtin_nontemporal_store(
        o, reinterpret_cast<v4f*>(reinterpret_cast<char*>(out) + (tid << 4)));
}

extern "C" void kernel_launch(void* const* d_in, const int* in_sizes, int n_in,
                              void* d_out, int out_size, void* d_ws, size_t ws_size,
                              hipStream_t stream) {
    (void)in_sizes; (void)n_in; (void)out_size; (void)d_ws; (void)ws_size;
    const int*   idx = (const int*)d_in[0];   // action: int32 [BATCH,1]
    const float* W   = (const float*)d_in[1]; // [VOCAB, EMBED] f32
    const float* b   = (const float*)d_in[2]; // [EMBED] f32
    float*       out = (float*)d_out;         // [BATCH, EMBED] f32

    const int threads = BATCH * 4;            // 4 lanes per embedding row
    const int block   = 256;                  // 8 wave32s per block
    const int grid    = threads / block;      // exact cover: 1024 blocks
    embed_gather_kernel<<<grid, block, 0, stream>>>(idx, W, b, out);
}